// M8BlankSlateCodebook_32899449487364
// MI455X (gfx1250) — compile-verified
//
#include <hip/hip_runtime.h>
#include <hip/hip_bf16.h>

typedef __attribute__((ext_vector_type(2))) float v2f;
typedef __attribute__((ext_vector_type(4))) float v4f;
typedef __attribute__((ext_vector_type(8))) float v8f;

#define B_ROWS  131072
#define HDIM    896
#define DDIM    16
#define KCODES  2000
#define KTILES  (KCODES / 16)      // 125
#define WPB     4                  // waves per block
#define ESTR    18                 // emb LDS stride (even + bank-conflict-free)
#define ZSTR    18                 // z_choked staging stride

__global__ __launch_bounds__(32 * WPB, 1)
void vq_fused_kernel(const float* __restrict__ z,
                     const float* __restrict__ Wd,     // (D,H)
                     const float* __restrict__ bdown,  // (D,)
                     const float* __restrict__ emb,    // (K,D)
                     float* __restrict__ zOut,         // (B,16)
                     int*   __restrict__ idxOut)       // (B,)
{
    __shared__ float embS[KCODES * ESTR];          // 144000 B
    __shared__ float biasS[KCODES];                //   8000 B
    __shared__ float zstage[WPB][16 * ZSTR];       //   4608 B
    __shared__ int   idxS[WPB][16];                //    256 B

    const int tid = threadIdx.x;

    // ---- stage codebook into LDS (padded, conflict-free stride) ----
    for (int i = tid; i < KCODES * DDIM; i += blockDim.x) {
        int k = i >> 4, d = i & 15;
        embS[k * ESTR + d] = emb[i];
    }
    __syncthreads();
    // ---- bias = -0.5 * ||e_k||^2 ----
    for (int k = tid; k < KCODES; k += blockDim.x) {
        float s = 0.f;
        #pragma unroll
        for (int d = 0; d < 16; ++d) { float e = embS[k * ESTR + d]; s += e * e; }
        biasS[k] = -0.5f * s;
    }
    __syncthreads();

    const int wave = tid >> 5;
    const int lane = tid & 31;
    const int lo   = lane & 15;     // row / col / N index within tile
    const int hi   = lane >> 4;     // which K-half of the 16x4 fragment
    const int tile = blockIdx.x * WPB + wave;   // 16-row tile id
    const int rowBase = tile * 16;

    // ================= GEMM1: z_choked = z @ Wd^T + b =================
    // A (16x4): lane lo holds row lo, K = hi*2 + {0,1}
    // B (4x16): lane lo holds col lo of Wd^T, i.e. Wd[lo][h + hi*2 + {0,1}]
    const float* zp = z  + (size_t)(rowBase + lo) * HDIM + hi * 2;
    const float* wp = Wd + (size_t)lo * HDIM + hi * 2;
    const float bval = bdown[lo];

    v8f acc[4];
    #pragma unroll
    for (int q = 0; q < 4; ++q)
        #pragma unroll
        for (int j = 0; j < 8; ++j) acc[q][j] = (q == 0) ? bval : 0.f;

    #pragma unroll 1
    for (int h = 0; h < HDIM; h += 16) {
        __builtin_prefetch(zp + h + 256, 0, 0);   // global_prefetch_b8 (speculative -> GL2)
        #pragma unroll
        for (int q = 0; q < 4; ++q) {
            v2f a = *(const v2f*)(zp + h + 4 * q);
            v2f b = *(const v2f*)(wp + h + 4 * q);
            acc[q] = __builtin_amdgcn_wmma_f32_16x16x4_f32(false, a, false, b,
                                                           (short)0, acc[q],
                                                           false, false);
        }
    }
    v8f zc = (acc[0] + acc[1]) + (acc[2] + acc[3]);
    // C-layout: zc[j] @ lane = z_choked[j + 8*hi][lo]

    // ---- C-layout -> A-layout transpose via per-wave LDS staging ----
    #pragma unroll
    for (int j = 0; j < 8; ++j)
        zstage[wave][(j + 8 * hi) * ZSTR + lo] = zc[j];
    asm volatile("s_wait_dscnt 0" ::: "memory");

    // A fragments for the distance GEMM: row = lo, K-chunk s, half hi
    v2f afrag[4];
    #pragma unroll
    for (int s = 0; s < 4; ++s)
        afrag[s] = *(const v2f*)&zstage[wave][lo * ZSTR + 4 * s + hi * 2];

    // ===== GEMM2 + argmax: score = z_choked . e_k - 0.5||e_k||^2 =====
    v8f best; int bidx[8];
    #pragma unroll
    for (int j = 0; j < 8; ++j) { best[j] = -3.402823466e38f; bidx[j] = 0; }

    auto do_tile = [&](int t) {
        const float bias = biasS[t * 16 + lo];
        v8f c;
        #pragma unroll
        for (int j = 0; j < 8; ++j) c[j] = bias;
        const float* ebase = &embS[(t * 16 + lo) * ESTR + hi * 2];
        #pragma unroll
        for (int s = 0; s < 4; ++s) {
            v2f bf = *(const v2f*)(ebase + 4 * s);
            c = __builtin_amdgcn_wmma_f32_16x16x4_f32(false, afrag[s], false, bf,
                                                      (short)0, c, false, false);
        }
        const int code = t * 16 + lo;
        #pragma unroll
        for (int j = 0; j < 8; ++j) {
            if (c[j] > best[j]) { best[j] = c[j]; bidx[j] = code; }
        }
    };

    // 124 tiles unrolled by 2 (two independent WMMA chains in flight), then tail
    #pragma unroll 1
    for (int t = 0; t < KTILES - 1; t += 2) {
        do_tile(t);
        do_tile(t + 1);
    }
    do_tile(KTILES - 1);

    // ---- cross-lane argmax within each 16-lane group (wave32) ----
    #pragma unroll
    for (int j = 0; j < 8; ++j) {
        float bv = best[j]; int bi = bidx[j];
        #pragma unroll
        for (int off = 8; off >= 1; off >>= 1) {
            float ov = __shfl_xor(bv, off, 16);
            int   oi = __shfl_xor(bi, off, 16);
            if (ov > bv || (ov == bv && oi < bi)) { bv = ov; bi = oi; }
        }
        if (lo == j) idxS[wave][j + 8 * hi] = bi;   // row j+8*hi
    }
    asm volatile("s_wait_dscnt 0" ::: "memory");

    // ---- outputs: idx + gathered codebook rows (z_st == emb[idx]) ----
    if (lane < 16)
        idxOut[rowBase + lane] = idxS[wave][lane];

    const int r    = lane >> 1;          // 32 lanes -> 16 rows x 2 halves
    const int half = lane & 1;
    const int code = idxS[wave][r];
    const float* ep = &embS[code * ESTR + half * 8];
    float* op = zOut + (size_t)(rowBase + r) * 16 + half * 8;
    v4f o0, o1;
    #pragma unroll
    for (int i = 0; i < 4; ++i) { o0[i] = ep[i]; o1[i] = ep[4 + i]; }
    *(v4f*)op       = o0;
    *((v4f*)op + 1) = o1;
}

extern "C" void kernel_launch(void* const* d_in, const int* in_sizes, int n_in,
                              void* d_out, int out_size, void* d_ws, size_t ws_size,
                              hipStream_t stream) {
    const float* z     = (const float*)d_in[0];
    const float* Wd    = (const float*)d_in[1];
    const float* bdown = (const float*)d_in[2];
    const float* emb   = (const float*)d_in[3];
    float* zOut  = (float*)d_out;
    int*   idxOut = (int*)(zOut + (size_t)B_ROWS * DDIM);

    dim3 grid(B_ROWS / 16 / WPB);   // 2048 blocks
    dim3 block(32 * WPB);           // 128 threads = 4 waves
    hipLaunchKernelGGL(vq_fused_kernel, grid, block, 0, stream,
                       z, Wd, bdown, emb, zOut, idxOut);
}